// SGLV_11708080849035
// MI455X (gfx1250) — compile-verified
//
#include <hip/hip_runtime.h>
#include <cstdint>

// ---------------------------------------------------------------------------
// NeRF-style ray marching on MI455X (gfx1250).
//  - Pre-pass: re-tile volume [11,128,128,128] -> [128,128,128,12] (48B recs)
//    so one corner fetch = 3 aligned b128 loads instead of 11 far-apart lines
//    (volume is L2-resident at 92-101 MB; this cuts L2 gather traffic ~4.5x).
//  - Main pass: 1 thread/ray, 100 samples, trilinear gather + compositing.
//    All corner addresses use uniform-SGPR-base + 32-bit voffset addressing.
//  - Ray data staged to LDS via gfx1250 async-to-LDS ops + s_wait_asynccnt.
// ---------------------------------------------------------------------------

static constexpr int   S_SAMPLES = 100;
static constexpr int   NGRID     = 128;
static constexpr int   NVOX      = NGRID * NGRID * NGRID;
static constexpr int   CPAD      = 12;            // 11 channels padded to 12
static constexpr float HALF_SIZE = 50.0f;         // SIZE/2

// ---- Pre-pass: channel-interleave the volume into workspace ---------------
__global__ __launch_bounds__(256) void vol_interleave_kernel(
    const float* __restrict__ vol, float* __restrict__ voli) {
    int i = blockIdx.x * blockDim.x + threadIdx.x;  // voxel id (z*128+y)*128+x
    if (i >= NVOX) return;
    float v[11];
#pragma unroll
    for (int c = 0; c < 11; ++c) v[c] = vol[(unsigned)(c * NVOX) + (unsigned)i];
    float4* dst = reinterpret_cast<float4*>(voli + (size_t)i * CPAD);
    dst[0] = make_float4(v[0], v[1], v[2], v[3]);
    dst[1] = make_float4(v[4], v[5], v[6], v[7]);
    dst[2] = make_float4(v[8], v[9], v[10], 0.0f);
}

// ---- Main ray-march kernel ------------------------------------------------
template <bool INTERLEAVED>
__global__ __launch_bounds__(256) void raymarch_kernel(
    const float* __restrict__ ro, const float* __restrict__ rd,
    const float* __restrict__ vol, float* __restrict__ out, int R) {
    __shared__ float sray[256 * 6];  // [0..767]=origins xyz, [768..1535]=dirs

    const int t = threadIdx.x;
    const int r = blockIdx.x * 256 + t;

    // Stage this block's ray data into LDS with gfx1250 async-to-LDS loads.
    {
        const uint64_t gro  = (uint64_t)ro + (uint64_t)blockIdx.x * 256u * 12u;
        const uint64_t grd  = (uint64_t)rd + (uint64_t)blockIdx.x * 256u * 12u;
        const unsigned gofs = (unsigned)t * 12u;                       // per-lane byte off
        const unsigned ldsO = (unsigned)(uintptr_t)(&sray[0]) + (unsigned)t * 12u;
        const unsigned ldsD = ldsO + 256u * 12u;
        if (r < R) {
            asm volatile("global_load_async_to_lds_b32 %0, %1, %2"
                         :: "v"(ldsO), "v"(gofs), "s"(gro) : "memory");
            asm volatile("global_load_async_to_lds_b32 %0, %1, %2 offset:4"
                         :: "v"(ldsO), "v"(gofs), "s"(gro) : "memory");
            asm volatile("global_load_async_to_lds_b32 %0, %1, %2 offset:8"
                         :: "v"(ldsO), "v"(gofs), "s"(gro) : "memory");
            asm volatile("global_load_async_to_lds_b32 %0, %1, %2"
                         :: "v"(ldsD), "v"(gofs), "s"(grd) : "memory");
            asm volatile("global_load_async_to_lds_b32 %0, %1, %2 offset:4"
                         :: "v"(ldsD), "v"(gofs), "s"(grd) : "memory");
            asm volatile("global_load_async_to_lds_b32 %0, %1, %2 offset:8"
                         :: "v"(ldsD), "v"(gofs), "s"(grd) : "memory");
        }
        asm volatile("s_wait_asynccnt 0x0" ::: "memory");
    }
    __syncthreads();
    if (r >= R) return;

    const float ox  = sray[t * 3 + 0];
    const float oy  = sray[t * 3 + 1];
    const float oz  = sray[t * 3 + 2];
    const float rdx = sray[768 + t * 3 + 0];
    const float rdy = sray[768 + t * 3 + 1];
    const float rdz = sray[768 + t * 3 + 2];

    // AABB entry/exit exactly as the reference computes it.
    const float ivx = 1.0f / rdx, ivy = 1.0f / rdy, ivz = 1.0f / rdz;
    const float tminx = (-HALF_SIZE - ox) * ivx, tmaxx = (HALF_SIZE - ox) * ivx;
    const float tminy = (-HALF_SIZE - oy) * ivy, tmaxy = (HALF_SIZE - oy) * ivy;
    const float tminz = (-HALF_SIZE - oz) * ivz, tmaxz = (HALF_SIZE - oz) * ivz;
    const float t0 = fmaxf(fmaxf(tminx, fmaxf(tminy, tminz)), 0.0f);
    const float t1 = fminf(tmaxx, fminf(tmaxy, tmaxz));
    const float dstep = (t1 - t0) * (1.0f / (float)(S_SAMPLES - 1));

    const char* __restrict__ volb = reinterpret_cast<const char*>(vol);

    float accc0 = 0.f, accc1 = 0.f, accc2 = 0.f;
    float accw0 = 0.f, accw1 = 0.f, accw2 = 0.f;
    float accs0 = 0.f, accs1 = 0.f, accs2 = 0.f;
    float accl = 0.f;
    float trans = 1.0f;

#pragma unroll 1
    for (int s = 0; s < S_SAMPLES; ++s) {
        const float tt = fmaf((float)s, dstep, t0);
        const float px = fmaf(tt, rdx, ox);
        const float py = fmaf(tt, rdy, oy);
        const float pz = fmaf(tt, rdz, oz);

        // grid coords in [-1,1] -> voxel coords [0,127]; clamp to avoid int UB,
        // clamped region is always fully out-of-bounds (weights become 0).
        float ix = (px * (1.0f / HALF_SIZE) + 1.0f) * (0.5f * (float)(NGRID - 1));
        float iy = (py * (1.0f / HALF_SIZE) + 1.0f) * (0.5f * (float)(NGRID - 1));
        float iz = (pz * (1.0f / HALF_SIZE) + 1.0f) * (0.5f * (float)(NGRID - 1));
        ix = fminf(fmaxf(ix, -16.0f), (float)(NGRID + 16));
        iy = fminf(fmaxf(iy, -16.0f), (float)(NGRID + 16));
        iz = fminf(fmaxf(iz, -16.0f), (float)(NGRID + 16));

        const float x0f = floorf(ix), y0f = floorf(iy), z0f = floorf(iz);
        const float fx = ix - x0f, fy = iy - y0f, fz = iz - z0f;
        const int x0 = (int)x0f, y0 = (int)y0f, z0 = (int)z0f;

        int   xcs[2], ycs[2], zcs[2];
        float wxs[2], wys[2], wzs[2];
        xcs[0] = min(max(x0, 0), NGRID - 1);
        xcs[1] = min(max(x0 + 1, 0), NGRID - 1);
        ycs[0] = min(max(y0, 0), NGRID - 1);
        ycs[1] = min(max(y0 + 1, 0), NGRID - 1);
        zcs[0] = min(max(z0, 0), NGRID - 1);
        zcs[1] = min(max(z0 + 1, 0), NGRID - 1);
        wxs[0] = (1.0f - fx) * ((x0 >= 0 && x0 < NGRID) ? 1.0f : 0.0f);
        wxs[1] = fx * ((x0 + 1 >= 0 && x0 + 1 < NGRID) ? 1.0f : 0.0f);
        wys[0] = (1.0f - fy) * ((y0 >= 0 && y0 < NGRID) ? 1.0f : 0.0f);
        wys[1] = fy * ((y0 + 1 >= 0 && y0 + 1 < NGRID) ? 1.0f : 0.0f);
        wzs[0] = (1.0f - fz) * ((z0 >= 0 && z0 < NGRID) ? 1.0f : 0.0f);
        wzs[1] = fz * ((z0 + 1 >= 0 && z0 + 1 < NGRID) ? 1.0f : 0.0f);

        float samp[11];
#pragma unroll
        for (int k = 0; k < 11; ++k) samp[k] = 0.0f;

#pragma unroll
        for (int dz = 0; dz < 2; ++dz) {
#pragma unroll
            for (int dy = 0; dy < 2; ++dy) {
                const float    wzy = wzs[dz] * wys[dy];
                const unsigned zy  = (unsigned)((zcs[dz] * NGRID + ycs[dy]) * NGRID);
#pragma unroll
                for (int dxc = 0; dxc < 2; ++dxc) {
                    const float    wgt  = wzy * wxs[dxc];
                    const unsigned vidx = zy + (unsigned)xcs[dxc];
                    if (INTERLEAVED) {
                        // 48-byte record, 16B aligned -> 3x global_load_b128 with
                        // 32-bit voffset addressing (base stays in SGPRs).
                        const unsigned boff = vidx * 48u;
                        const float4   A = *reinterpret_cast<const float4*>(volb + boff);
                        const float4   B = *reinterpret_cast<const float4*>(volb + boff + 16u);
                        const float4   C = *reinterpret_cast<const float4*>(volb + boff + 32u);
                        samp[0]  = fmaf(wgt, A.x, samp[0]);
                        samp[1]  = fmaf(wgt, A.y, samp[1]);
                        samp[2]  = fmaf(wgt, A.z, samp[2]);
                        samp[3]  = fmaf(wgt, A.w, samp[3]);
                        samp[4]  = fmaf(wgt, B.x, samp[4]);
                        samp[5]  = fmaf(wgt, B.y, samp[5]);
                        samp[6]  = fmaf(wgt, B.z, samp[6]);
                        samp[7]  = fmaf(wgt, B.w, samp[7]);
                        samp[8]  = fmaf(wgt, C.x, samp[8]);
                        samp[9]  = fmaf(wgt, C.y, samp[9]);
                        samp[10] = fmaf(wgt, C.z, samp[10]);
                    } else {
#pragma unroll
                        for (int c = 0; c < 11; ++c) {
                            const unsigned boff = ((unsigned)(c * NVOX) + vidx) * 4u;
                            const float v = *reinterpret_cast<const float*>(volb + boff);
                            samp[c] = fmaf(wgt, v, samp[c]);
                        }
                    }
                }
            }
        }

        // Compositing: inclusive cumprod, wts = alpha * trans.
        const float alpha = samp[3];
        trans = trans * (1.0f - alpha + 1e-10f);
        const float wq = alpha * trans;
        accc0 = fmaf(wq, samp[0], accc0);
        accc1 = fmaf(wq, samp[1], accc1);
        accc2 = fmaf(wq, samp[2], accc2);
        accw0 = fmaf(wq, samp[4], accw0);
        accw1 = fmaf(wq, samp[5], accw1);
        accw2 = fmaf(wq, samp[6], accw2);
        accl  = fmaf(wq, samp[7], accl);
        accs0 = fmaf(wq, samp[8], accs0);
        accs1 = fmaf(wq, samp[9], accs1);
        accs2 = fmaf(wq, samp[10], accs2);
    }

    const float dot = accs0 * rdx + accs1 * rdy + accs2 * rdz;
    const float e   = expf(accl * (dot - 1.0f));
    out[(size_t)r * 3 + 0] = fmaf(accw0, e, accc0);
    out[(size_t)r * 3 + 1] = fmaf(accw1, e, accc1);
    out[(size_t)r * 3 + 2] = fmaf(accw2, e, accc2);
}

// ---------------------------------------------------------------------------
extern "C" void kernel_launch(void* const* d_in, const int* in_sizes, int n_in,
                              void* d_out, int out_size, void* d_ws, size_t ws_size,
                              hipStream_t stream) {
    (void)n_in; (void)out_size;
    const float* ro  = (const float*)d_in[0];
    const float* rd  = (const float*)d_in[1];
    const float* vol = (const float*)d_in[2];
    float* out = (float*)d_out;
    const int R = in_sizes[0] / 3;

    const size_t need    = (size_t)NVOX * CPAD * sizeof(float);  // ~100.7 MB
    const int    rblocks = (R + 255) / 256;

    if (ws_size >= need) {
        float* voli = (float*)d_ws;
        vol_interleave_kernel<<<(NVOX + 255) / 256, 256, 0, stream>>>(vol, voli);
        raymarch_kernel<true><<<rblocks, 256, 0, stream>>>(ro, rd, voli, out, R);
    } else {
        raymarch_kernel<false><<<rblocks, 256, 0, stream>>>(ro, rd, vol, out, R);
    }
}